// MultiProbeAttentionPooler_5901285065163
// MI455X (gfx1250) — compile-verified
//
#include <hip/hip_runtime.h>
#include <math.h>

#define B_ 16
#define T_ 4096
#define D_ 1024
#define P_ 16

typedef float v2f __attribute__((ext_vector_type(2)));
typedef float v4f __attribute__((ext_vector_type(4)));
typedef float v8f __attribute__((ext_vector_type(8)));

// monotonic uint key for float max via integer atomics (deterministic, always available)
__device__ __forceinline__ unsigned f2key(float f) {
  unsigned u = __float_as_uint(f);
  return ((int)u >= 0) ? (u ^ 0x80000000u) : ~u;
}
__device__ __forceinline__ float key2f(unsigned k) {
  unsigned u = (k & 0x80000000u) ? (k ^ 0x80000000u) : ~k;
  return __uint_as_float(u);
}

__global__ void mpap_init_kernel(unsigned* __restrict__ ws_max) {
  ws_max[threadIdx.x] = 0u;  // 0 < key(-inf): identity for key-space max
}

// Phase 1: logits[b][p][t] = x[b][t]·W[p] + bias[p], written transposed into the
// attention output region; also per-(b,p) running max via atomic key-max.
__global__ void __launch_bounds__(128)
mpap_logits_kernel(const float* __restrict__ x, const float* __restrict__ W,
                   const float* __restrict__ bias, float* __restrict__ att,
                   unsigned* __restrict__ ws_max) {
  __shared__ float ldsL[16 * 156];  // [p][t_local], pad 156 -> conflict-free
  const int lane = threadIdx.x & 31;
  const int wave = threadIdx.x >> 5;
  const int lo = lane & 15;
  const int hi = lane >> 4;
  const int b = blockIdx.y;
  const int tblk = blockIdx.x * 128;
  const int trow = tblk + wave * 32 + lo;

  // A layout (16x4 f32): lane lo = row M, VGPR j holds K = 2*hi + j  -> float2 load
  const float* xr0 = x + ((size_t)b * T_ + trow) * D_ + hi * 2;
  const float* xr1 = xr0 + 16 * D_;
  // B layout (4x16 f32): lane lo = col N, VGPR j holds K = 2*hi + j; B[k][p] = W[p][k0+k]
  const float* wr = W + lo * D_ + hi * 2;

  v8f c0 = {}; v8f c1 = {};
  for (int k0 = 0; k0 < D_; k0 += 64) {
    if (k0 + 256 < D_) {
      __builtin_prefetch(xr0 + k0 + 256, 0, 0);
      __builtin_prefetch(xr1 + k0 + 256, 0, 0);
    }
#pragma unroll
    for (int kk = 0; kk < 64; kk += 4) {
      v2f a0 = *(const v2f*)(xr0 + k0 + kk);
      v2f a1 = *(const v2f*)(xr1 + k0 + kk);
      v2f bf = *(const v2f*)(wr + k0 + kk);
      c0 = __builtin_amdgcn_wmma_f32_16x16x4_f32(false, a0, false, bf, (short)0, c0, false, false);
      c1 = __builtin_amdgcn_wmma_f32_16x16x4_f32(false, a1, false, bf, (short)0, c1, false, false);
    }
  }

  // C layout: VGPR r, lane: token_local M = r + 8*hi, probe N = lo
  const float bb = bias[lo];
  float m = -3.402823466e38f;
#pragma unroll
  for (int r = 0; r < 8; ++r) {
    c0[r] += bb; c1[r] += bb;
    m = fmaxf(m, fmaxf(c0[r], c1[r]));
  }
#pragma unroll
  for (int r = 0; r < 8; ++r) {
    const int tl = wave * 32 + r + hi * 8;
    ldsL[lo * 156 + tl]      = c0[r];
    ldsL[lo * 156 + tl + 16] = c1[r];
  }
  m = fmaxf(m, __shfl_xor(m, 16, 32));  // lanes L and L^16 share probe lo
  if (lane < 16) atomicMax(&ws_max[b * 16 + lo], f2key(m));
  __syncthreads();

  // coalesced b128 stores: att[b][p][tblk + 0..127]
#pragma unroll
  for (int rep = 0; rep < 4; ++rep) {
    const int f4 = rep * 128 + (int)threadIdx.x;  // 512 float4
    const int p = f4 >> 5;
    const int tq = f4 & 31;
    v4f v = *(const v4f*)&ldsL[p * 156 + tq * 4];
    *(v4f*)(att + ((size_t)b * 16 + p) * T_ + tblk + tq * 4) = v;
  }
}

// Phase 2: per-(b,p) softmax over T, fully in one block (deterministic tree sum),
// normalizes attention in place.
__global__ void __launch_bounds__(256)
mpap_softmax_kernel(float* __restrict__ att, const unsigned* __restrict__ ws_max) {
  __shared__ float red[8];
  float* row = att + (size_t)blockIdx.x * T_;
  const float m = key2f(ws_max[blockIdx.x]);
  v4f v[4];
  float s = 0.f;
#pragma unroll
  for (int rep = 0; rep < 4; ++rep) {
    v4f t = *(const v4f*)(row + (rep * 256 + threadIdx.x) * 4);
#pragma unroll
    for (int q = 0; q < 4; ++q) { t[q] = __expf(t[q] - m); s += t[q]; }
    v[rep] = t;
  }
#pragma unroll
  for (int off = 16; off > 0; off >>= 1) s += __shfl_xor(s, off, 32);
  if ((threadIdx.x & 31) == 0) red[threadIdx.x >> 5] = s;
  __syncthreads();
  if (threadIdx.x == 0) {
    float tot = 0.f;
#pragma unroll
    for (int w = 0; w < 8; ++w) tot += red[w];
    red[0] = 1.0f / tot;
  }
  __syncthreads();
  const float inv = red[0];
#pragma unroll
  for (int rep = 0; rep < 4; ++rep) {
    v4f t = v[rep];
#pragma unroll
    for (int q = 0; q < 4; ++q) t[q] *= inv;
    *(v4f*)(row + (rep * 256 + threadIdx.x) * 4) = t;
  }
}

// Phase 3: pooled[b][p][d] = sum_t att[b][p][t] * x[b][t][d]  (M=P=16, N=16 d-cols,
// K=T split over 4 waves, LDS-reduced for determinism)
__global__ void __launch_bounds__(128)
mpap_pool_kernel(const float* __restrict__ x, const float* __restrict__ att,
                 float* __restrict__ pooled) {
  __shared__ float ldsC[4 * 256];
  const int lane = threadIdx.x & 31;
  const int wave = threadIdx.x >> 5;
  const int lo = lane & 15;
  const int hi = lane >> 4;
  const int b = blockIdx.y;
  const int d0 = blockIdx.x * 16;

  const float* arow = att + ((size_t)b * 16 + lo) * T_ + hi * 2;  // A = attn^T [p][t]
  const float* xcol = x + (size_t)b * T_ * D_ + d0 + lo;          // B = x [t][d]

  v8f ca = {}; v8f cb = {};
  const int tbeg = wave * (T_ / 4);
  const int tend = tbeg + (T_ / 4);
  for (int t0 = tbeg; t0 < tend; t0 += 16) {
    if (t0 + 64 < tend) __builtin_prefetch(xcol + (size_t)(t0 + 64) * D_, 0, 0);
#pragma unroll
    for (int tt = 0; tt < 16; tt += 4) {
      const int t = t0 + tt;
      v2f a = *(const v2f*)(arow + t);
      v2f bf;
      bf.x = xcol[(size_t)(t + hi * 2) * D_];
      bf.y = xcol[(size_t)(t + hi * 2 + 1) * D_];
      if ((tt & 4) == 0)
        ca = __builtin_amdgcn_wmma_f32_16x16x4_f32(false, a, false, bf, (short)0, ca, false, false);
      else
        cb = __builtin_amdgcn_wmma_f32_16x16x4_f32(false, a, false, bf, (short)0, cb, false, false);
    }
  }
  // C layout: VGPR r, lane: p = r + 8*hi, d_local = lo
#pragma unroll
  for (int r = 0; r < 8; ++r)
    ldsC[wave * 256 + (r + 8 * hi) * 16 + lo] = ca[r] + cb[r];
  __syncthreads();
#pragma unroll
  for (int e0 = 0; e0 < 256; e0 += 128) {
    const int e = e0 + (int)threadIdx.x;
    const float sum = ldsC[e] + ldsC[256 + e] + ldsC[512 + e] + ldsC[768 + e];
    const int p = e >> 4, n = e & 15;
    pooled[((size_t)b * 16 + p) * D_ + d0 + n] = sum;
  }
}

extern "C" void kernel_launch(void* const* d_in, const int* in_sizes, int n_in,
                              void* d_out, int out_size, void* d_ws, size_t ws_size,
                              hipStream_t stream) {
  const float* x = (const float*)d_in[0];
  const float* W = (const float*)d_in[1];
  const float* bias = (const float*)d_in[2];
  float* pooled = (float*)d_out;                       // [B,P,D]
  float* att = (float*)d_out + (size_t)B_ * P_ * D_;   // [B,P,T]
  unsigned* ws_max = (unsigned*)d_ws;                  // 256 keys

  mpap_init_kernel<<<1, B_ * P_, 0, stream>>>(ws_max);
  mpap_logits_kernel<<<dim3(T_ / 128, B_), 128, 0, stream>>>(x, W, bias, att, ws_max);
  mpap_softmax_kernel<<<B_ * P_, 256, 0, stream>>>(att, ws_max);
  mpap_pool_kernel<<<dim3(D_ / 16, B_), 128, 0, stream>>>(x, att, pooled);
}